// Hadamard_layer_16140487098685
// MI455X (gfx1250) — compile-verified
//
#include <hip/hip_runtime.h>

// CDNA5 / gfx1250: wave32, WMMA 16x16 tiles.
typedef float v2f __attribute__((ext_vector_type(2)));
typedef float v8f __attribute__((ext_vector_type(8)));

#define NVEC   4096          // 2^12 state length
#define NBATCH 256

// One workgroup transforms one 4096-float vector (a given batch index, real or
// imag plane) via the factorization H4096 = H16 (x) H16 (x) H16:
//   three axis contractions of the [16,16,16] tensor with the constant H16,
//   each done as 16 WMMA tiles (one per wave), K=16 via 4 chained
//   v_wmma_f32_16x16x4_f32 ops. H16 entries (+-0.25) are synthesized in
//   registers from popcount — the 67 MB H input is never read.
__global__ __launch_bounds__(512)
void hadamard_fwht_wmma_kernel(const float* __restrict__ state_real,
                               const float* __restrict__ state_imag,
                               float* __restrict__ out) {
    __shared__ float lds[NVEC];

    const int v = blockIdx.x;            // 0..511
    const int c = v & 1;                 // 0 = real plane, 1 = imag plane
    const int b = v >> 1;                // batch index
    const float* __restrict__ src = (c ? state_imag : state_real) + (size_t)b * NVEC;
    float* __restrict__ dst = out + (size_t)c * (NBATCH * NVEC) + (size_t)b * NVEC;

    const int t = threadIdx.x;           // 0..511

    // ---- cooperative load: 4096 floats, 8 per thread via two b128 loads ----
    {
        const float4* s4 = (const float4*)src;
        float4* l4 = (float4*)lds;
        l4[t]       = s4[t];
        l4[t + 512] = s4[t + 512];
    }
    __syncthreads();

    const int lane = t & 31;
    const int wave = t >> 5;             // 0..15 : tile index f
    const int m16  = lane & 15;          // A: row M ; B: col N ; D: col N
    const int hi   = lane >> 4;          // K-half selector (0 or 1)
    const int f    = wave;

    // ---- constant B fragments: B[k][n] = H16[kc*4 + k][n], same all stages.
    // Lane layout (mirror of documented A 16x4 layout; H16 symmetric, so a
    // k<->n transposed interpretation yields identical register contents):
    //   lane: n = lane%16, k = (lane/16)*2 + vgpr.
    v2f bfrag[4];
#pragma unroll
    for (int kc = 0; kc < 4; ++kc) {
#pragma unroll
        for (int vv = 0; vv < 2; ++vv) {
            const int j = kc * 4 + hi * 2 + vv;
            bfrag[kc][vv] = (__popc(j & m16) & 1) ? -0.25f : 0.25f;
        }
    }

    // Stage address maps over the [16,16,16] tensor (strides 256,16,1):
    //  stage0: read f*256 + M*16 + j   ; write f*256 + M*16  + N
    //  stage1: read f*256 + j*16 + M   ; write f*256 + N*16  + M
    //  stage2: read j*256 + f*16 + M   ; write N*256 + f*16  + M
    // Tile read/write sets are disjoint across waves -> in-place per stage.
#pragma unroll
    for (int stage = 0; stage < 3; ++stage) {
        int s_f, s_m, s_j, so_f, so_m, so_n;
        if (stage == 0)      { s_f = 256; s_m = 16; s_j = 1;   so_f = 256; so_m = 16; so_n = 1;   }
        else if (stage == 1) { s_f = 256; s_m = 1;  s_j = 16;  so_f = 256; so_m = 1;  so_n = 16;  }
        else                 { s_f = 16;  s_m = 1;  s_j = 256; so_f = 16;  so_m = 1;  so_n = 256; }

        // A fragments (documented f32 16x4 layout):
        //   lane: M = lane%16, k = (lane/16)*2 + vgpr, global j = kc*4 + k.
        v2f afrag[4];
#pragma unroll
        for (int kc = 0; kc < 4; ++kc) {
#pragma unroll
            for (int vv = 0; vv < 2; ++vv) {
                const int j = kc * 4 + hi * 2 + vv;
                afrag[kc][vv] = lds[f * s_f + m16 * s_m + j * s_j];
            }
        }

        // D = sum_kc A_kc x B_kc  (K = 16 total), fp32 throughout.
        v8f acc = {};
#pragma unroll
        for (int kc = 0; kc < 4; ++kc) {
            acc = __builtin_amdgcn_wmma_f32_16x16x4_f32(
                /*neg_a=*/false, afrag[kc],
                /*neg_b=*/false, bfrag[kc],
                /*c_mod=*/(short)0, acc,
                /*reuse_a=*/false, /*reuse_b=*/false);
        }

        // Scatter D back (documented C/D layout: vgpr g, lane -> row M =
        // (lane/16)*8 + g, col N = lane%16).
#pragma unroll
        for (int g = 0; g < 8; ++g) {
            const int M = hi * 8 + g;
            lds[f * so_f + M * so_m + m16 * so_n] = acc[g];
        }
        __syncthreads();
    }

    // ---- cooperative store ----
    {
        const float4* l4 = (const float4*)lds;
        float4* d4 = (float4*)dst;
        d4[t]       = l4[t];
        d4[t + 512] = l4[t + 512];
    }
}

extern "C" void kernel_launch(void* const* d_in, const int* in_sizes, int n_in,
                              void* d_out, int out_size, void* d_ws, size_t ws_size,
                              hipStream_t stream) {
    const float* state_real = (const float*)d_in[0];   // [256, 4096]
    const float* state_imag = (const float*)d_in[1];   // [256, 4096]
    // d_in[2] is H [4096,4096]; unused — the transform is synthesized from the
    // Hadamard tensor structure (exact same values, fp32 math).
    float* out = (float*)d_out;                        // [2, 256, 4096] (real, imag)

    (void)in_sizes; (void)n_in; (void)out_size; (void)d_ws; (void)ws_size;

    dim3 grid(2 * NBATCH);   // 512 vectors (real+imag planes)
    dim3 block(512);         // 16 waves -> 16 WMMA tiles per stage
    hadamard_fwht_wmma_kernel<<<grid, block, 0, stream>>>(state_real, state_imag, out);
}